// HIDecoder_84464826843593
// MI455X (gfx1250) — compile-verified
//
#include <hip/hip_runtime.h>

// ---------------------------------------------------------------------------
// Fused HIDecoder forward for gfx1250 (MI455X).
//   y = z @ Wy^T + by  (fp32 WMMA 16x16x4), reshaped to gamma[B,T,Y]
//   lin_m/lin_v = einsum('bty,tdy->btd') + bias
//   Gaussian log-prob + masked split; outputs concat(log_p_x, log_p_x_missing,
//   mean_x, var_x) in fp32.
// One wave owns 16 batch rows; per t it computes the 16x32 gamma tile with
// two WMMA accumulators (y=0..15 and y=16..31), stages it in padded LDS,
// and finishes the head math in-register. Intermediate y never hits HBM.
// ---------------------------------------------------------------------------

typedef __attribute__((ext_vector_type(2))) float v2f;
typedef __attribute__((ext_vector_type(8))) float v8f;

#define B_DIM 32768
#define Z_DIM 256
#define Y_DIM 32
#define T_DIM 64
#define D_DIM 4
#define EPS_F 1e-3f
#define LOG2PI_F 1.8378770664093453f

__device__ __forceinline__ float softplus_stable(float x) {
    // log1p(exp(x)) computed stably (matches jax.nn.softplus)
    return fmaxf(x, 0.0f) + log1pf(expf(-fabsf(x)));
}

__global__ __launch_bounds__(256, 1) void hidecoder_fused_kernel(
    const float* __restrict__ z,          // [B, Z]
    const float* __restrict__ batch_x,    // [B, T, D]
    const int*   __restrict__ miss_list,  // [B, T]
    const float* __restrict__ data_mean,  // [T, D]
    const float* __restrict__ data_var,   // [T, D]
    const float* __restrict__ Wy,         // [T*Y, Z]
    const float* __restrict__ by,         // [T*Y]
    const float* __restrict__ Wm,         // [T, D, Y]
    const float* __restrict__ bm,         // [T, D]
    const float* __restrict__ Wv,         // [T, D, Y]
    const float* __restrict__ bv,         // [T, D]
    float* __restrict__ out)              // concat outputs
{
    // Per-wave gamma staging: 16 rows x 32 y, padded to 33 floats/row
    // (stride 33 is coprime with 64 banks -> conflict-free).
    __shared__ float gamma_s[8 * 16 * 33];

    const int lane = threadIdx.x & 31;
    const int w    = threadIdx.x >> 5;   // wave id in block (0..7)
    const int lo   = lane & 15;
    const int hi   = lane >> 4;          // 0 or 1

    // This wave's 16 batch rows: blockIdx.x*128 + w*16 + [0..15]
    const int row = blockIdx.x * 128 + w * 16 + lo;

    // ---- Stage A fragments (z rows) in registers, WMMA 16x4xF32 A-layout:
    //   VGPR0: lanes 0-15 K=4j+0, lanes 16-31 K=4j+2
    //   VGPR1: lanes 0-15 K=4j+1, lanes 16-31 K=4j+3
    // => lane loads float2 at z[row, 4j + 2*hi]. Read once, reused 128x.
    v2f a[64];
    {
        const float* zp = z + (size_t)row * Z_DIM + 2 * hi;
        #pragma unroll
        for (int j = 0; j < 64; ++j) a[j] = *(const v2f*)(zp + 4 * j);
    }

    float* gs = gamma_s + w * (16 * 33);

    const size_t o1 = (size_t)B_DIM * T_DIM;                      // log_p_x_missing
    const size_t o2 = 2 * o1;                                     // mean_x
    const size_t o3 = o2 + (size_t)B_DIM * T_DIM * (size_t)D_DIM; // var_x

    const int m  = lo;        // row-in-tile this lane finishes
    const int db = 2 * hi;    // this lane handles d = db, db+1
    const int bb = blockIdx.x * 128 + w * 16 + m;

    for (int t = 0; t < T_DIM; ++t) {
        // Prefetch next t's Wy tile (emits global_prefetch_b8; line-granular).
        if (t + 1 < T_DIM) {
            __builtin_prefetch(Wy + ((size_t)((t + 1) * Y_DIM + lane)) * Z_DIM, 0, 1);
        }

        // ---- WMMA: gamma[16,32] = z_tile[16,256] @ Wy[t*32 .. t*32+31, :]^T
        // B[k,n] = Wy[n0+n, k]; B-layout mirrors A with M->N:
        //   lane loads float2 at Wy[n0 + lane%16 (+16), 4j + 2*hi].
        const float* wy0 = Wy + ((size_t)(t * Y_DIM + lo)) * Z_DIM + 2 * hi;
        const float* wy1 = wy0 + 16 * (size_t)Z_DIM;

        v8f acc0 = {0.f, 0.f, 0.f, 0.f, 0.f, 0.f, 0.f, 0.f}; // y = 0..15
        v8f acc1 = {0.f, 0.f, 0.f, 0.f, 0.f, 0.f, 0.f, 0.f}; // y = 16..31

        #pragma unroll
        for (int j = 0; j < 64; ++j) {
            v2f fb0 = *(const v2f*)(wy0 + 4 * j);
            v2f fb1 = *(const v2f*)(wy1 + 4 * j);
            acc0 = __builtin_amdgcn_wmma_f32_16x16x4_f32(
                false, a[j], false, fb0, (short)0, acc0, false, false);
            acc1 = __builtin_amdgcn_wmma_f32_16x16x4_f32(
                false, a[j], false, fb1, (short)0, acc1, false, false);
        }

        // Bias: gamma[m, y] += by[t*32 + y]; in C/D layout y = lane%16 (+16).
        const float bias0 = by[t * Y_DIM + lo];
        const float bias1 = by[t * Y_DIM + 16 + lo];

        // C/D layout: VGPR r, lane L -> M = r + 8*(L/16), N = L%16.
        #pragma unroll
        for (int r = 0; r < 8; ++r) {
            const int mm = r + 8 * hi;
            gs[mm * 33 + lo]      = acc0[r] + bias0;
            gs[mm * 33 + 16 + lo] = acc1[r] + bias1;
        }
        // Same-wave DS ops are in-order; each wave reads only its own region,
        // so no block barrier is required before the loads below.

        // ---- Head: lin = gamma @ W[t]^T + b  for d = db, db+1
        float lm0 = bm[t * D_DIM + db], lm1 = bm[t * D_DIM + db + 1];
        float lv0 = bv[t * D_DIM + db], lv1 = bv[t * D_DIM + db + 1];
        const float* wm0 = Wm + ((size_t)(t * D_DIM + db)) * Y_DIM;
        const float* wm1 = wm0 + Y_DIM;
        const float* wv0 = Wv + ((size_t)(t * D_DIM + db)) * Y_DIM;
        const float* wv1 = wv0 + Y_DIM;

        #pragma unroll
        for (int y = 0; y < Y_DIM; ++y) {
            const float g = gs[m * 33 + y];
            lm0 = fmaf(g, wm0[y], lm0);
            lm1 = fmaf(g, wm1[y], lm1);
            lv0 = fmaf(g, wv0[y], lv0);
            lv1 = fmaf(g, wv1[y], lv1);
        }

        // ---- Gaussian log-likelihood pieces
        const float dv0 = fmaxf(data_var[t * D_DIM + db],     EPS_F);
        const float dv1 = fmaxf(data_var[t * D_DIM + db + 1], EPS_F);
        const float ev0 = fminf(fmaxf(softplus_stable(lv0), EPS_F), 1e20f);
        const float ev1 = fminf(fmaxf(softplus_stable(lv1), EPS_F), 1e20f);
        const float mx0 = sqrtf(dv0) * lm0 + data_mean[t * D_DIM + db];
        const float mx1 = sqrtf(dv1) * lm1 + data_mean[t * D_DIM + db + 1];
        const float vx0 = dv0 * ev0;
        const float vx1 = dv1 * ev1;

        const size_t bt = (size_t)bb * T_DIM + t;
        const v2f xv = *(const v2f*)(batch_x + bt * D_DIM + db);
        const float e0 = xv[0] - mx0;
        const float e1 = xv[1] - mx1;

        float term = e0 * e0 / vx0 + logf(vx0)
                   + e1 * e1 / vx1 + logf(vx1)
                   + 2.0f * LOG2PI_F;

        // Store mean_x / var_x (this lane's two consecutive d's).
        v2f mv; mv[0] = mx0; mv[1] = mx1;
        v2f vv; vv[0] = vx0; vv[1] = vx1;
        *(v2f*)(out + o2 + bt * D_DIM + db) = mv;
        *(v2f*)(out + o3 + bt * D_DIM + db) = vv;

        // Sum the two d-pairs across the half-wave split (lane ^ 16).
        term += __shfl_xor(term, 16, 32);
        const float logp = -0.5f * term;

        if (hi == 0) {
            const float mk = (float)miss_list[bt];
            out[bt]      = logp * mk;
            out[o1 + bt] = logp * (1.0f - mk);
        }
    }
}

extern "C" void kernel_launch(void* const* d_in, const int* in_sizes, int n_in,
                              void* d_out, int out_size, void* d_ws, size_t ws_size,
                              hipStream_t stream) {
    const float* z         = (const float*)d_in[0];
    const float* batch_x   = (const float*)d_in[1];
    const int*   miss_list = (const int*)  d_in[2];
    const float* data_mean = (const float*)d_in[3];
    const float* data_var  = (const float*)d_in[4];
    const float* Wy        = (const float*)d_in[5];
    const float* by        = (const float*)d_in[6];
    const float* Wm        = (const float*)d_in[7];
    const float* bm        = (const float*)d_in[8];
    const float* Wv        = (const float*)d_in[9];
    const float* bv        = (const float*)d_in[10];

    // 256 blocks x 256 threads: 8 waves/block, 16 batch rows/wave -> 32768 rows.
    hidecoder_fused_kernel<<<dim3(B_DIM / 128), dim3(256), 0, stream>>>(
        z, batch_x, miss_list, data_mean, data_var,
        Wy, by, Wm, bm, Wv, bv, (float*)d_out);
}